// LlamaMoE_30425548325398
// MI455X (gfx1250) — compile-verified
//
#include <hip/hip_runtime.h>

#define H_DIM  1024
#define I_DIM  2816
#define I2_DIM 5632
#define E_NUM  8
#define T_NUM  2048
#define XS_STRIDE 1028   // 1028 % 64 == 4 -> conflict-free across 16 lanes
#define AS_STRIDE 2820   // 2820 % 64 == 4

typedef __attribute__((ext_vector_type(2))) float v2f;
typedef __attribute__((ext_vector_type(8))) float v8f;

__device__ __forceinline__ float fast_silu(float g) {
    // g * sigmoid(g); denominator in [1, inf) -> fast v_rcp_f32 is safe
    return g * __builtin_amdgcn_rcpf(1.f + __expf(-g));
}

__global__ void moe_zero_kernel(float* __restrict__ out, int n4) {
    float4 z = make_float4(0.f, 0.f, 0.f, 0.f);
    for (int i = blockIdx.x * blockDim.x + threadIdx.x; i < n4;
         i += gridDim.x * blockDim.x)
        ((float4*)out)[i] = z;
}

__global__ void moe_router_kernel(const float* __restrict__ x,
                                  const float* __restrict__ gate_w,
                                  float* __restrict__ cw) {
    const int wv   = threadIdx.x >> 5;
    const int lane = threadIdx.x & 31;
    const int t    = blockIdx.x * 8 + wv;

    const float* xr = x + (size_t)t * H_DIM;
    float acc[E_NUM];
#pragma unroll
    for (int e = 0; e < E_NUM; ++e) acc[e] = 0.f;

    for (int h = lane; h < H_DIM; h += 32) {
        float xv = xr[h];
        const float4* gw = (const float4*)(gate_w + (size_t)h * E_NUM);
        float4 g0 = gw[0], g1 = gw[1];
        acc[0] += xv * g0.x; acc[1] += xv * g0.y;
        acc[2] += xv * g0.z; acc[3] += xv * g0.w;
        acc[4] += xv * g1.x; acc[5] += xv * g1.y;
        acc[6] += xv * g1.z; acc[7] += xv * g1.w;
    }
#pragma unroll
    for (int e = 0; e < E_NUM; ++e) {
#pragma unroll
        for (int off = 16; off >= 1; off >>= 1)
            acc[e] += __shfl_xor(acc[e], off, 32);
    }

    // softmax (fp32) -> top-2 -> renormalize. softmax is monotone; the
    // renormalized top-2 ratio p_i/(p_i+p_j) is scale-invariant.
    float m = acc[0];
#pragma unroll
    for (int e = 1; e < E_NUM; ++e) m = fmaxf(m, acc[e]);
    float p[E_NUM];
#pragma unroll
    for (int e = 0; e < E_NUM; ++e) p[e] = __expf(acc[e] - m);

    int i1 = 0;
#pragma unroll
    for (int e = 1; e < E_NUM; ++e) if (p[e] > p[i1]) i1 = e;
    int i2 = (i1 == 0) ? 1 : 0;
#pragma unroll
    for (int e = 0; e < E_NUM; ++e)
        if (e != i1 && p[e] > p[i2]) i2 = e;

    float denom = p[i1] + p[i2];   // exact fp32 divide here (tiny, precision-relevant)
    if (lane < E_NUM) {
        float w = 0.f;
        if (lane == i1) w = p[i1] / denom;
        else if (lane == i2) w = p[i2] / denom;
        cw[(size_t)t * E_NUM + lane] = w;
    }
}

__launch_bounds__(256, 1)
__global__ void moe_mlp_kernel(const float* __restrict__ x,
                               const float* __restrict__ base_wgu,
                               const float* __restrict__ base_wd,
                               const float* __restrict__ exp_wgu,
                               const float* __restrict__ exp_wd,
                               const float* __restrict__ cw,
                               float* __restrict__ out) {
    __shared__ float xs[16 * XS_STRIDE];   // 16 x 1024 x-tile (padded)
    __shared__ float as_[16 * AS_STRIDE];  // 16 x 2816 activation tile (padded)

    const int tile = blockIdx.x;
    const int z    = blockIdx.z;          // 0 = base, 1..8 = experts
    const int row0 = tile * 16;

    const float* wgu;
    const float* wd;
    if (z == 0) {
        wgu = base_wgu; wd = base_wd;
    } else {
        wgu = exp_wgu + (size_t)(z - 1) * H_DIM * I2_DIM;
        wd  = exp_wd  + (size_t)(z - 1) * I_DIM * H_DIM;
        // uniform early-exit if no token in this tile routes to expert z-1
        float mx = 0.f;
#pragma unroll
        for (int r = 0; r < 16; ++r)
            mx = fmaxf(mx, cw[(size_t)(row0 + r) * E_NUM + (z - 1)]);
        if (mx == 0.f) return;
    }

    // ---- stage x tile into LDS (row-major, padded stride) ----
    {
        const float4* src = (const float4*)(x + (size_t)row0 * H_DIM);
        for (int i = threadIdx.x; i < 16 * (H_DIM / 4); i += 256) {
            int r = i >> 8;          // 256 float4 per row
            int c = i & 255;
            ((float4*)(xs + r * XS_STRIDE))[c] = src[(size_t)r * (H_DIM / 4) + c];
        }
    }
    __syncthreads();

    const int wv   = threadIdx.x >> 5;
    const int lane = threadIdx.x & 31;
    const int hi   = lane >> 4;   // K-half select for A/B fragments
    const int lm   = lane & 15;   // row (A) / column (B,C)

    // ---- phase 1: gu = x @ wgu ; act = silu(gate) * up -> LDS ----
    // Each wave handles a PAIR of adjacent 16-col tiles: one A fragment feeds
    // 4 WMMAs (gate0, up0, gate1, up1); B loads cover 128B/half-wave.
    for (int pt = wv; pt < (I_DIM / 32); pt += 8) {
        const int c0 = pt * 32;
        v8f accg0 = {}, accu0 = {}, accg1 = {}, accu1 = {};
        const float* bg = wgu + (size_t)c0 + lm;            // gate cols
        const float* bu = wgu + (size_t)(c0 + I_DIM) + lm;  // up cols
        const float* ar = xs + lm * XS_STRIDE;
        for (int k0 = 0; k0 < H_DIM; k0 += 4) {
            const int kk = k0 + 2 * hi;
            const size_t r0 = (size_t)kk * I2_DIM, r1 = (size_t)(kk + 1) * I2_DIM;
            v2f a;  a.x  = ar[kk];       a.y  = ar[kk + 1];
            v2f g0; g0.x = bg[r0];       g0.y = bg[r1];
            v2f g1; g1.x = bg[r0 + 16];  g1.y = bg[r1 + 16];
            v2f u0; u0.x = bu[r0];       u0.y = bu[r1];
            v2f u1; u1.x = bu[r0 + 16];  u1.y = bu[r1 + 16];
            accg0 = __builtin_amdgcn_wmma_f32_16x16x4_f32(
                false, a, false, g0, (short)0, accg0, false, false);
            accu0 = __builtin_amdgcn_wmma_f32_16x16x4_f32(
                false, a, false, u0, (short)0, accu0, false, false);
            accg1 = __builtin_amdgcn_wmma_f32_16x16x4_f32(
                false, a, false, g1, (short)0, accg1, false, false);
            accu1 = __builtin_amdgcn_wmma_f32_16x16x4_f32(
                false, a, false, u1, (short)0, accu1, false, false);
        }
#pragma unroll
        for (int i = 0; i < 8; ++i) {
            const int row = i + 8 * hi;
            as_[row * AS_STRIDE + c0 + lm]      = fast_silu(accg0[i]) * accu0[i];
            as_[row * AS_STRIDE + c0 + 16 + lm] = fast_silu(accg1[i]) * accu1[i];
        }
    }
    __syncthreads();

    // per-row combine weights for this z-slice
    float sc[8];
#pragma unroll
    for (int i = 0; i < 8; ++i) {
        int row = row0 + i + 8 * hi;
        sc[i] = (z == 0) ? 1.f : cw[(size_t)row * E_NUM + (z - 1)];
    }

    // ---- phase 2: out += sc * (act @ wd) ----  (paired N-tiles)
    for (int pt = wv; pt < (H_DIM / 32); pt += 8) {
        const int n0 = pt * 32;
        v8f acc0 = {}, acc1 = {};
        const float* bp = wd + (size_t)n0 + lm;
        const float* ar = as_ + lm * AS_STRIDE;
        for (int k0 = 0; k0 < I_DIM; k0 += 4) {
            const int kk = k0 + 2 * hi;
            const size_t r0 = (size_t)kk * H_DIM, r1 = (size_t)(kk + 1) * H_DIM;
            v2f a;  a.x  = ar[kk];      a.y  = ar[kk + 1];
            v2f b0; b0.x = bp[r0];      b0.y = bp[r1];
            v2f b1; b1.x = bp[r0 + 16]; b1.y = bp[r1 + 16];
            acc0 = __builtin_amdgcn_wmma_f32_16x16x4_f32(
                false, a, false, b0, (short)0, acc0, false, false);
            acc1 = __builtin_amdgcn_wmma_f32_16x16x4_f32(
                false, a, false, b1, (short)0, acc1, false, false);
        }
#pragma unroll
        for (int i = 0; i < 8; ++i) {
            const int row = row0 + i + 8 * hi;
            float* orow = out + (size_t)row * H_DIM + n0 + lm;
            unsafeAtomicAdd(orow,      sc[i] * acc0[i]);
            unsafeAtomicAdd(orow + 16, sc[i] * acc1[i]);
        }
    }
}

extern "C" void kernel_launch(void* const* d_in, const int* in_sizes, int n_in,
                              void* d_out, int out_size, void* d_ws, size_t ws_size,
                              hipStream_t stream) {
    (void)in_sizes; (void)n_in; (void)ws_size;
    const float* x        = (const float*)d_in[0];
    const float* gate_w   = (const float*)d_in[1];
    const float* base_wgu = (const float*)d_in[2];
    const float* base_wd  = (const float*)d_in[3];
    const float* exp_wgu  = (const float*)d_in[4];
    const float* exp_wd   = (const float*)d_in[5];
    float* out = (float*)d_out;
    float* cw  = (float*)d_ws;   // T x 8 routing weights (64 KB)

    moe_zero_kernel<<<dim3(256), dim3(256), 0, stream>>>(out, out_size / 4);
    moe_router_kernel<<<dim3(T_NUM / 8), dim3(256), 0, stream>>>(x, gate_w, cw);
    moe_mlp_kernel<<<dim3(T_NUM / 16, 1, E_NUM + 1), dim3(256), 0, stream>>>(
        x, base_wgu, base_wd, exp_wgu, exp_wd, cw, out);
}